// RGCN_62251255989021
// MI455X (gfx1250) — compile-verified
//
#include <hip/hip_runtime.h>

typedef __attribute__((ext_vector_type(2))) float v2f;
typedef __attribute__((ext_vector_type(8))) float v8f;

#define RGCN_DIN 64  // both layers have 64-wide input

// ---------------------------------------------------------------------------
// Transform: for blockIdx.y = r < R: hw[r, n0:n0+16, :] = A_tile @ W[r]
//            for blockIdx.y = R (self): selfdst[n0:n0+16, :] = A_tile @ Wself + b
// One wave per 16-column output tile; K=64 loop of V_WMMA_F32_16X16X4_F32.
// ---------------------------------------------------------------------------
template <int DOUT>
__global__ __launch_bounds__(32 * (DOUT / 16))
void rgcn_transform(const float* __restrict__ h,      // [N, DIN]
                    const float* __restrict__ W,      // [R, DIN, DOUT]
                    const float* __restrict__ Wself,  // [DIN, DOUT]
                    const float* __restrict__ bias,   // [DOUT]
                    float* __restrict__ hw,           // [R, N, DOUT]
                    float* __restrict__ selfdst,      // [N, DOUT]
                    int N, int R) {
  __shared__ float Atile[16 * RGCN_DIN];

  const int n0  = blockIdx.x * 16;
  const int r   = blockIdx.y;  // 0..R-1 = relations, R = self-loop
  const int tid = threadIdx.x;

  // Stage 16 x 64 fp32 A tile (contiguous 4KB) into LDS, coalesced.
  const int gmax = N * RGCN_DIN;
  for (int i = tid; i < 16 * RGCN_DIN; i += blockDim.x) {
    const int g = n0 * RGCN_DIN + i;
    Atile[i] = (g < gmax) ? h[g] : 0.0f;
  }
  __syncthreads();

  const int lane = tid & 31;
  const int wave = tid >> 5;
  const int half = lane >> 4;   // 0: lanes 0-15, 1: lanes 16-31
  const int l15  = lane & 15;
  const int c0   = wave * 16;   // this wave's output-column tile

  const float* __restrict__ Wr =
      (r < R) ? (W + (size_t)r * RGCN_DIN * DOUT) : Wself;

  v8f acc = {};
#pragma unroll
  for (int k0 = 0; k0 < RGCN_DIN; k0 += 4) {
    const int ka = k0 + 2 * half;
    // A 16x4: lane = row M, half-wave selects K pair {ka, ka+1}. 8B LDS load.
    v2f a = *(const v2f*)&Atile[l15 * RGCN_DIN + ka];
    // B 4x16: lane = col N, half-wave selects K pair {ka, ka+1}.
    v2f b;
    b.x = Wr[(ka + 0) * DOUT + c0 + l15];
    b.y = Wr[(ka + 1) * DOUT + c0 + l15];
    acc = __builtin_amdgcn_wmma_f32_16x16x4_f32(
        /*neg_a=*/false, a, /*neg_b=*/false, b,
        /*c_mod=*/(short)0, acc, /*reuse_a=*/false, /*reuse_b=*/false);
  }

  // D 16x16 f32: VGPR v -> M = v (lanes 0-15) / v+8 (lanes 16-31), N = lane&15.
  const int col = c0 + l15;
#pragma unroll
  for (int v = 0; v < 8; ++v) {
    const int row = n0 + v + half * 8;
    if (row < N) {
      if (r < R)
        hw[((size_t)r * N + row) * DOUT + col] = acc[v];
      else
        selfdst[(size_t)row * DOUT + col] = acc[v] + bias[col];
    }
  }
}

// ---------------------------------------------------------------------------
// Scatter-add: out[dst[e], c] += hw[etype[e], src[e], c]  (segment_sum)
// thread = (edge, column); coalesced 256B/128B gather rows, coalesced f32
// atomics on the destination row.
// ---------------------------------------------------------------------------
template <int DOUT, int LOG2>
__global__ __launch_bounds__(256)
void rgcn_scatter(const float* __restrict__ hw,  // [R, N, DOUT]
                  const int* __restrict__ etypes,
                  const int* __restrict__ srcv,
                  const int* __restrict__ dstv,
                  float* __restrict__ out,       // [N, DOUT]
                  int N, long long total) {
  const long long idx = (long long)blockIdx.x * blockDim.x + threadIdx.x;
  if (idx >= total) return;
  const int e = (int)(idx >> LOG2);
  const int c = (int)(idx & (DOUT - 1));
  const float v = hw[((size_t)etypes[e] * N + srcv[e]) * DOUT + c];
  unsafeAtomicAdd(&out[(size_t)dstv[e] * DOUT + c], v);  // global_atomic_add_f32
}

__global__ __launch_bounds__(256) void rgcn_relu(float* __restrict__ x, int n) {
  const int i = blockIdx.x * blockDim.x + threadIdx.x;
  if (i < n) x[i] = fmaxf(x[i], 0.0f);
}

// ---------------------------------------------------------------------------
extern "C" void kernel_launch(void* const* d_in, const int* in_sizes, int n_in,
                              void* d_out, int out_size, void* d_ws,
                              size_t ws_size, hipStream_t stream) {
  const float* feat   = (const float*)d_in[0];  // [N, 64]
  const int*   srcv   = (const int*)d_in[1];    // [E]
  const int*   dstv   = (const int*)d_in[2];    // [E]
  const int*   etypes = (const int*)d_in[3];    // [E]
  const float* W1     = (const float*)d_in[4];  // [R, 64, 64]
  const float* Wself1 = (const float*)d_in[5];  // [64, 64]
  const float* b1     = (const float*)d_in[6];  // [64]
  const float* W2     = (const float*)d_in[7];  // [R, 64, 32]
  const float* Wself2 = (const float*)d_in[8];  // [64, 32]
  const float* b2     = (const float*)d_in[9];  // [32]
  float*       out    = (float*)d_out;          // [N, 32]

  const int N = in_sizes[0] / 64;
  const int E = in_sizes[1];
  const int R = in_sizes[4] / (64 * 64);

  // Workspace: [hw: R*N*64 f32 (reused as R*N*32 for layer 2)][x: N*64 f32]
  float* hw = (float*)d_ws;
  float* x  = (float*)((char*)d_ws + (size_t)R * N * 64 * sizeof(float));

  const int ntiles = (N + 15) / 16;

  // ---- Layer 1: x = relu(scatter(hw1) + feat @ Wself1 + b1) ----
  rgcn_transform<64><<<dim3(ntiles, R + 1), 128, 0, stream>>>(
      feat, W1, Wself1, b1, hw, x, N, R);

  const long long tot1 = (long long)E * 64;
  rgcn_scatter<64, 6><<<(unsigned)((tot1 + 255) / 256), 256, 0, stream>>>(
      hw, etypes, srcv, dstv, x, N, tot1);

  rgcn_relu<<<(N * 64 + 255) / 256, 256, 0, stream>>>(x, N * 64);

  // ---- Layer 2: out = scatter(hw2) + x @ Wself2 + b2 ----
  rgcn_transform<32><<<dim3(ntiles, R + 1), 64, 0, stream>>>(
      x, W2, Wself2, b2, hw, out, N, R);

  const long long tot2 = (long long)E * 32;
  rgcn_scatter<32, 5><<<(unsigned)((tot2 + 255) / 256), 256, 0, stream>>>(
      hw, etypes, srcv, dstv, out, N, tot2);
}